// UniversalHomogeneousSAGEModel_87033217286400
// MI455X (gfx1250) — compile-verified
//
#include <hip/hip_runtime.h>
#include <hip/hip_bf16.h>
#include <stdint.h>

typedef __attribute__((ext_vector_type(2))) float v2f;
typedef __attribute__((ext_vector_type(8))) float v8f;

static constexpr int kC   = 128;  // feature channels
static constexpr int kOut = 64;   // head output channels

static inline int cdiv(long long a, long long b) { return (int)((a + b - 1) / b); }

// ---------------------------------------------------------------------------
// CDNA5 async global->LDS copy (per-lane 16B), tracked by ASYNCcnt.
// lds_off: 32-bit LDS byte address (low 32 bits of the flat LDS pointer are
// the wave-relative LDS offset per the CDNA5 aperture mapping).
// ---------------------------------------------------------------------------
__device__ __forceinline__ void async_g2l_b128(uint32_t lds_off, const void* gptr) {
  asm volatile("global_load_async_to_lds_b128 %0, %1, off"
               :
               : "v"(lds_off), "v"((uint64_t)(uintptr_t)gptr)
               : "memory");
}

__device__ __forceinline__ void wait_asynccnt0() {
  asm volatile("s_wait_asynccnt 0x0" ::: "memory");
}

// ---------------------------------------------------------------------------
// 1) in-degree via float atomics (edge-parallel)
// ---------------------------------------------------------------------------
__global__ void k_degree(const long long* __restrict__ ei, float* __restrict__ deg, int E) {
  int e = blockIdx.x * blockDim.x + threadIdx.x;
  if (e < E) {
    int dst = (int)ei[(size_t)E + e];
    atomicAdd(&deg[dst], 1.0f);
  }
}

__global__ void k_invdeg(float* __restrict__ deg, int N) {
  int n = blockIdx.x * blockDim.x + threadIdx.x;
  if (n < N) deg[n] = 1.0f / fmaxf(deg[n], 1.0f);
}

// ---------------------------------------------------------------------------
// 2) neighbor scatter with mean folded in: agg[dst] += x[src] * invdeg[dst].
//    One wave per edge, 4 channels per lane: a contiguous 512B read of x[src]
//    and contiguous global_atomic_add_f32 into agg[dst].
// ---------------------------------------------------------------------------
__global__ void k_scatter(const float* __restrict__ x, const long long* __restrict__ ei,
                          const float* __restrict__ invdeg, float* __restrict__ agg, int E) {
  long long t = (long long)blockIdx.x * blockDim.x + threadIdx.x;
  int e = (int)(t >> 5);
  if (e >= E) return;
  int c4 = (int)(t & 31) * 4;  // 4 channels per lane, 32 lanes = 128 channels
  long long src = ei[e];
  long long dst = ei[(size_t)E + e];
  float w = invdeg[dst];  // broadcast within the wave
  float4 v = *(const float4*)(x + src * kC + c4);
  float* a = agg + dst * kC + c4;
  atomicAdd(a + 0, v.x * w);
  atomicAdd(a + 1, v.y * w);
  atomicAdd(a + 2, v.z * w);
  atomicAdd(a + 3, v.w * w);
}

// ---------------------------------------------------------------------------
// 3) fused SAGE conv: h = agg@Wn^T + x@Wr^T + b ; x' = relu(LN(h))
//    Block: 16 nodes x 128 out-channels. A-tiles DMA'd into LDS via async
//    loads; 8 waves, each one 16x16 tile, 64 x V_WMMA_F32_16X16X4_F32.
// ---------------------------------------------------------------------------
__global__ __launch_bounds__(256) void k_sage_conv(
    const float* __restrict__ xin, const float* __restrict__ agg,
    const float* __restrict__ Wn, const float* __restrict__ Wr,
    const float* __restrict__ bias, const float* __restrict__ gamma,
    const float* __restrict__ beta, float* __restrict__ xout, int N) {
  __shared__ __align__(16) float sA[16][kC];      // agg tile (already mean-scaled)
  __shared__ __align__(16) float sX[16][kC];      // x tile
  __shared__ __align__(16) float sH[16][kC + 4];  // h tile (padded)

  const int tid = threadIdx.x;
  const int n0 = blockIdx.x * 16;

  // async DMA staging: 2 tiles x 8KB; each thread moves 2 x 16B per tile
  {
    const float4* gA = (const float4*)(agg + (size_t)n0 * kC);
    const float4* gX = (const float4*)(xin + (size_t)n0 * kC);
    const uint32_t lA = (uint32_t)(uintptr_t)&sA[0][0];
    const uint32_t lX = (uint32_t)(uintptr_t)&sX[0][0];
#pragma unroll
    for (int i = 0; i < 2; ++i) {
      const int q = tid + i * 256;  // float4 index 0..511
      async_g2l_b128(lA + q * 16, gA + q);
      async_g2l_b128(lX + q * 16, gX + q);
    }
    wait_asynccnt0();
  }
  __syncthreads();

  const int wave = tid >> 5;
  const int lane = tid & 31;
  const int col0 = wave * 16;    // 16 output channels per wave
  const int r = lane & 15;       // A row / B column
  const int hi = lane >> 4;      // K-pair select per ISA layout

  const float* wn_row = Wn + (size_t)(col0 + r) * kC;  // B = W^T: column j == row j of W
  const float* wr_row = Wr + (size_t)(col0 + r) * kC;

  v8f acc = {};
#pragma unroll
  for (int kk = 0; kk < kC; kk += 4) {
    const int k2 = kk + 2 * hi;
    v2f aA = *(const v2f*)&sA[r][k2];
    v2f bN = *(const v2f*)&wn_row[k2];
    acc = __builtin_amdgcn_wmma_f32_16x16x4_f32(false, aA, false, bN, (short)0, acc, false, false);
    v2f aX = *(const v2f*)&sX[r][k2];
    v2f bR = *(const v2f*)&wr_row[k2];
    acc = __builtin_amdgcn_wmma_f32_16x16x4_f32(false, aX, false, bR, (short)0, acc, false, false);
  }

  // bias + spill h to LDS (C/D layout: VGPR i -> row i (+8 for hi lanes), col = lane%16)
  const float bj = bias[col0 + r];
#pragma unroll
  for (int i = 0; i < 8; ++i) {
    sH[i + hi * 8][col0 + r] = acc[i] + bj;
  }
  __syncthreads();

  // LayerNorm + ReLU: 16 threads per node, 8 channels each, shuffle reduce
  const int node = tid >> 4;
  const int cb = (tid & 15) * 8;
  float v[8];
  float s = 0.f, ss = 0.f;
#pragma unroll
  for (int i = 0; i < 8; ++i) {
    float t = sH[node][cb + i];
    v[i] = t;
    s += t;
    ss += t * t;
  }
#pragma unroll
  for (int m = 1; m < 16; m <<= 1) {
    s += __shfl_xor(s, m, 32);
    ss += __shfl_xor(ss, m, 32);
  }
  const float mu = s * (1.0f / kC);
  const float var = fmaxf(ss * (1.0f / kC) - mu * mu, 0.0f);
  const float rstd = rsqrtf(var + 1e-5f);
  float o[8];
#pragma unroll
  for (int i = 0; i < 8; ++i) {
    float t = (v[i] - mu) * rstd * gamma[cb + i] + beta[cb + i];
    o[i] = fmaxf(t, 0.0f);
  }
  float* op = xout + (size_t)(n0 + node) * kC + cb;
  *(float4*)(op) = make_float4(o[0], o[1], o[2], o[3]);
  *(float4*)(op + 4) = make_float4(o[4], o[5], o[6], o[7]);
}

// ---------------------------------------------------------------------------
// 4) head: out[0:M] = x @ W_head^T + b_head   (M=1024, OUT=64, K=128)
//    Block: 32 rows x 64 cols = 8 waves of 16x16 tiles; async LDS staging.
// ---------------------------------------------------------------------------
__global__ __launch_bounds__(256) void k_head(const float* __restrict__ x,
                                              const float* __restrict__ Wh,
                                              const float* __restrict__ bh,
                                              float* __restrict__ out, int M) {
  __shared__ __align__(16) float sX[32][kC];
  const int tid = threadIdx.x;
  const int m0 = blockIdx.x * 32;
  {
    const float4* g = (const float4*)(x + (size_t)m0 * kC);
    const uint32_t l = (uint32_t)(uintptr_t)&sX[0][0];
#pragma unroll
    for (int i = 0; i < 4; ++i) {  // 32*128 floats = 1024 float4
      const int q = tid + i * 256;
      async_g2l_b128(l + q * 16, g + q);
    }
    wait_asynccnt0();
  }
  __syncthreads();

  const int wave = tid >> 5;
  const int lane = tid & 31;
  const int mt = wave >> 2;      // 2 row tiles
  const int nt = wave & 3;       // 4 col tiles
  const int rb = mt * 16;
  const int col0 = nt * 16;
  const int r = lane & 15;
  const int hi = lane >> 4;

  const float* w_row = Wh + (size_t)(col0 + r) * kC;
  v8f acc = {};
#pragma unroll
  for (int kk = 0; kk < kC; kk += 4) {
    const int k2 = kk + 2 * hi;
    v2f a = *(const v2f*)&sX[rb + r][k2];
    v2f b = *(const v2f*)&w_row[k2];
    acc = __builtin_amdgcn_wmma_f32_16x16x4_f32(false, a, false, b, (short)0, acc, false, false);
  }
  const float bj = bh[col0 + r];
#pragma unroll
  for (int i = 0; i < 8; ++i) {
    int m = m0 + rb + i + hi * 8;
    out[(size_t)m * kOut + col0 + r] = acc[i] + bj;
  }
}

// ---------------------------------------------------------------------------
extern "C" void kernel_launch(void* const* d_in, const int* in_sizes, int n_in,
                              void* d_out, int out_size, void* d_ws, size_t ws_size,
                              hipStream_t stream) {
  const float* x_in      = (const float*)d_in[0];
  const long long* ei    = (const long long*)d_in[1];
  const float* Wn        = (const float*)d_in[2];
  const float* Wr        = (const float*)d_in[3];
  const float* bconv     = (const float*)d_in[4];
  const float* gamma     = (const float*)d_in[5];
  const float* beta      = (const float*)d_in[6];
  const float* Wh        = (const float*)d_in[7];
  const float* bh        = (const float*)d_in[8];

  const int N = in_sizes[0] / kC;
  const int E = in_sizes[1] / 2;
  const int L = in_sizes[2] / (kC * kC);
  const int M = out_size / kOut;  // num_seed

  // workspace layout: deg | agg | xbuf
  char* ws = (char*)d_ws;
  size_t off = 0;
  float* deg = (float*)(ws + off);
  off += (((size_t)N * 4) + 255) & ~(size_t)255;
  float* agg = (float*)(ws + off);
  off += (((size_t)N * kC * 4) + 255) & ~(size_t)255;
  float* xbuf = (float*)(ws + off);

  // degrees (once)
  hipMemsetAsync(deg, 0, (size_t)N * 4, stream);
  k_degree<<<cdiv(E, 256), 256, 0, stream>>>(ei, deg, E);
  k_invdeg<<<cdiv(N, 256), 256, 0, stream>>>(deg, N);

  const float* xcur = x_in;
  for (int l = 0; l < L; ++l) {
    hipMemsetAsync(agg, 0, (size_t)N * kC * 4, stream);
    k_scatter<<<cdiv((long long)E * 32, 256), 256, 0, stream>>>(xcur, ei, deg, agg, E);
    k_sage_conv<<<N / 16, 256, 0, stream>>>(
        xcur, agg, Wn + (size_t)l * kC * kC, Wr + (size_t)l * kC * kC,
        bconv + (size_t)l * kC, gamma + (size_t)l * kC, beta + (size_t)l * kC, xbuf, N);
    xcur = xbuf;  // conv is in-place-safe per 16-row block
  }

  k_head<<<M / 32, 256, 0, stream>>>(xcur, Wh, bh, (float*)d_out, M);
}